// NGCF_62191126446184
// MI455X (gfx1250) — compile-verified
//
#include <hip/hip_runtime.h>
#include <hip/hip_bf16.h>

#define N_USERS  100000
#define M_ITEMS  200000
#define N_NODES  (N_USERS + M_ITEMS)     // 300000
#define N_EDGES  1200000
#define D        64
#define N_LAYERS 3
#define BATCH    4096
#define NTILES   (N_NODES / 16)          // 18750 (divisible)
#define PSTRIDE  80                      // padded float2 stride per K-pair row (bank-conflict free)

typedef float v2f __attribute__((ext_vector_type(2)));
typedef float v8f __attribute__((ext_vector_type(8)));

// ---------------------------------------------------------------- zero (float4)
__global__ __launch_bounds__(256) void k_zero4(float4* __restrict__ p, int n4) {
    int i = blockIdx.x * 256 + threadIdx.x;
    if (i < n4) p[i] = make_float4(0.f, 0.f, 0.f, 0.f);
}

// ---------------------------------------------------------------- init ego = concat(user_emb, item_emb)
__global__ __launch_bounds__(256) void k_init(const float* __restrict__ ue,
                                              const float* __restrict__ ie,
                                              float* __restrict__ ego) {
    int i = blockIdx.x * 256 + threadIdx.x;
    if (i < N_USERS * D)           ego[i] = ue[i];
    else if (i < N_NODES * D)      ego[i] = ie[i - N_USERS * D];
}

// ---------------------------------------------------------------- SpMM scatter: side[rows[e]] += vals[e]*ego[cols[e]]
// 16 threads per edge, float4 per thread. ego/side are L2-resident (77MB each < 192MB L2),
// so both the gather and the fp32 hardware atomics run at L2, not HBM, speed.
__global__ __launch_bounds__(256) void k_spmm(const int* __restrict__ rows,
                                              const int* __restrict__ cols,
                                              const float* __restrict__ vals,
                                              const float* __restrict__ x,
                                              float* __restrict__ out) {
    int t = blockIdx.x * 256 + threadIdx.x;
    int e = t >> 4;
    if (e >= N_EDGES) return;
    int q = (t & 15) * 4;
    float v = vals[e];
    int cs = cols[e];
    int rd = rows[e];
    const float4 xv = *(const float4*)(x + (size_t)cs * D + q);
    float* o = out + (size_t)rd * D + q;
    unsafeAtomicAdd(o + 0, v * xv.x);
    unsafeAtomicAdd(o + 1, v * xv.y);
    unsafeAtomicAdd(o + 2, v * xv.z);
    unsafeAtomicAdd(o + 3, v * xv.w);
}

// ---------------------------------------------------------------- fused transform:
// ego_new = normalize(leaky_relu(side@Wgc + bgc + (ego*side)@Wbi + bbi))   (in-place on ego)
// One wave per 16-row tile; V_WMMA_F32_16X16X4_F32, K=64 in 16 steps, 4 column tiles.
// Weights pre-swizzled in LDS to K-pair layout so each B operand is one aligned ds_load_b64:
//   sWp[m][p*PSTRIDE + col] = (W[2p][col], W[2p+1][col])
// Lane needs pair p = kk/2 + hi  ->  b = (W[kk+2hi][col], W[kk+2hi+1][col]), exactly the
// ISA B-matrix layout (VGPR0 = rows K+0/K+2, VGPR1 = rows K+1/K+3 across lane halves).
// PSTRIDE=80: lanes 16-31 sit +160 dwords from lanes 0-15 -> banks 32..63 vs 0..31, conflict-free.
__global__ __launch_bounds__(256) void k_transform(const float* __restrict__ side,
                                                   float* __restrict__ ego,
                                                   const float* __restrict__ Wgc,
                                                   const float* __restrict__ bgc,
                                                   const float* __restrict__ Wbi,
                                                   const float* __restrict__ bbi) {
    __shared__ float2 sWp[2][32 * PSTRIDE];   // 40 KB
    __shared__ float  sB[2][D];

    const int tid = threadIdx.x;
    for (int i = tid; i < 2 * 2048; i += 256) {
        const int m   = i >> 11;
        const int j   = i & 2047;
        const int p   = j >> 6;
        const int col = j & 63;
        const float* W = m ? Wbi : Wgc;
        sWp[m][p * PSTRIDE + col] = make_float2(W[(2 * p) * D + col], W[(2 * p + 1) * D + col]);
    }
    if (tid < D) { sB[0][tid] = bgc[tid]; sB[1][tid] = bbi[tid]; }
    __syncthreads();

    const int wave = tid >> 5;
    const int lane = tid & 31;
    const int tile = blockIdx.x * 8 + wave;
    if (tile >= NTILES) return;           // wave-uniform: EXEC stays all-ones for WMMA

    const int r0    = tile * 16;
    const int mrow  = lane & 15;          // A-matrix row handled by this lane
    const int hi    = lane >> 4;          // 0: K+{0,1}, 1: K+{2,3}  (ISA 32-bit A layout)
    const int ncol  = lane & 15;
    const float* aS = side + (size_t)(r0 + mrow) * D;
    const float* aE = ego  + (size_t)(r0 + mrow) * D;

    v8f c[4];
    const v8f vzero = {0.f, 0.f, 0.f, 0.f, 0.f, 0.f, 0.f, 0.f};
    c[0] = vzero; c[1] = vzero; c[2] = vzero; c[3] = vzero;

    for (int kk = 0; kk < D; kk += 4) {
        const int acol = kk + hi * 2;
        const v2f sv = *(const v2f*)(aS + acol);
        const v2f ev = *(const v2f*)(aE + acol);
        const v2f aSide = sv;                 // A for side@Wgc
        v2f aBi; aBi.x = sv.x * ev.x; aBi.y = sv.y * ev.y;   // A for (ego*side)@Wbi

        const int p = (kk >> 1) + hi;         // K-pair index for this lane half

        // 4 consecutive WMMAs share the same A operand registers
        #pragma unroll
        for (int g = 0; g < 4; ++g) {
            const v2f bGc = *(const v2f*)&sWp[0][p * PSTRIDE + g * 16 + ncol];
            c[g] = __builtin_amdgcn_wmma_f32_16x16x4_f32(false, aSide, false, bGc,
                                                         (short)0, c[g], false, false);
        }
        #pragma unroll
        for (int g = 0; g < 4; ++g) {
            const v2f bBi = *(const v2f*)&sWp[1][p * PSTRIDE + g * 16 + ncol];
            c[g] = __builtin_amdgcn_wmma_f32_16x16x4_f32(false, aBi, false, bBi,
                                                         (short)0, c[g], false, false);
        }
    }

    // bias + leaky_relu(0.2)
    #pragma unroll
    for (int g = 0; g < 4; ++g) {
        const int col = g * 16 + ncol;
        const float bias = sB[0][col] + sB[1][col];
        #pragma unroll
        for (int r = 0; r < 8; ++r) {
            float x = c[g][r] + bias;
            c[g][r] = x > 0.f ? x : 0.2f * x;
        }
    }

    // per-row L2 normalize.  C element (g, r, lane): row = r + hi*8, col = g*16 + (lane&15).
    float scale[8];
    #pragma unroll
    for (int r = 0; r < 8; ++r) {
        float ss = c[0][r] * c[0][r] + c[1][r] * c[1][r] + c[2][r] * c[2][r] + c[3][r] * c[3][r];
        ss += __shfl_xor(ss, 1, 32);
        ss += __shfl_xor(ss, 2, 32);
        ss += __shfl_xor(ss, 4, 32);
        ss += __shfl_xor(ss, 8, 32);   // halves stay independent: rows r and r+8
        scale[r] = 1.0f / fmaxf(sqrtf(ss), 1e-12f);
    }

    #pragma unroll
    for (int g = 0; g < 4; ++g) {
        const int col = g * 16 + ncol;
        #pragma unroll
        for (int r = 0; r < 8; ++r) {
            const int row = r + hi * 8;
            ego[(size_t)(r0 + row) * D + col] = c[g][r] * scale[r];
        }
    }
}

// ---------------------------------------------------------------- accumulate selected rows into batch accumulators
__global__ __launch_bounds__(256) void k_accum(const float* __restrict__ ego,
                                               const int* __restrict__ users,
                                               const int* __restrict__ items,
                                               float* __restrict__ uacc,
                                               float* __restrict__ iacc) {
    int t = blockIdx.x * 256 + threadIdx.x;
    int b = t >> 6, d = t & 63;
    if (b < BATCH) {
        uacc[(size_t)b * D + d] += ego[(size_t)users[b] * D + d];
    } else if (b < 2 * BATCH) {
        int bb = b - BATCH;
        iacc[(size_t)bb * D + d] += ego[((size_t)N_USERS + items[bb]) * D + d];
    }
}

// ---------------------------------------------------------------- final: out[b] = dot(mean_u, mean_i) = dot(sum_u,sum_i)/16
__global__ __launch_bounds__(256) void k_dot(const float* __restrict__ uacc,
                                             const float* __restrict__ iacc,
                                             float* __restrict__ out) {
    int wave = (blockIdx.x * 256 + threadIdx.x) >> 5;
    int lane = threadIdx.x & 31;
    if (wave >= BATCH) return;
    const float* u = uacc + (size_t)wave * D;
    const float* i = iacc + (size_t)wave * D;
    float s = u[lane] * i[lane] + u[lane + 32] * i[lane + 32];
    s += __shfl_xor(s, 16, 32);
    s += __shfl_xor(s, 8, 32);
    s += __shfl_xor(s, 4, 32);
    s += __shfl_xor(s, 2, 32);
    s += __shfl_xor(s, 1, 32);
    if (lane == 0) out[wave] = s * 0.0625f;   // (1/4)*(1/4)
}

// ================================================================ host
extern "C" void kernel_launch(void* const* d_in, const int* in_sizes, int n_in,
                              void* d_out, int out_size, void* d_ws, size_t ws_size,
                              hipStream_t stream) {
    const int*   users    = (const int*)  d_in[0];
    const int*   items    = (const int*)  d_in[1];
    const int*   rows     = (const int*)  d_in[2];
    const int*   cols     = (const int*)  d_in[3];
    const float* vals     = (const float*)d_in[4];
    const float* user_emb = (const float*)d_in[5];
    const float* item_emb = (const float*)d_in[6];
    const float* W_gc     = (const float*)d_in[7];   // [3,64,64]
    const float* b_gc     = (const float*)d_in[8];   // [3,64]
    const float* W_bi     = (const float*)d_in[9];
    const float* b_bi     = (const float*)d_in[10];
    float*       out      = (float*)d_out;

    float* ego  = (float*)d_ws;                      // [N_NODES, 64]  76.8 MB
    float* side = ego  + (size_t)N_NODES * D;        // [N_NODES, 64]  76.8 MB
    float* uacc = side + (size_t)N_NODES * D;        // [BATCH, 64]
    float* iacc = uacc + (size_t)BATCH * D;          // [BATCH, 64]

    const int ND = N_NODES * D;                      // 19.2M

    k_init<<<(ND + 255) / 256, 256, 0, stream>>>(user_emb, item_emb, ego);
    k_zero4<<<(2 * BATCH * D / 4 + 255) / 256, 256, 0, stream>>>((float4*)uacc, 2 * BATCH * D / 4);
    k_accum<<<(2 * BATCH * D + 255) / 256, 256, 0, stream>>>(ego, users, items, uacc, iacc);

    for (int k = 0; k < N_LAYERS; ++k) {
        k_zero4<<<(ND / 4 + 255) / 256, 256, 0, stream>>>((float4*)side, ND / 4);
        k_spmm<<<(N_EDGES * 16 + 255) / 256, 256, 0, stream>>>(rows, cols, vals, ego, side);
        k_transform<<<(NTILES + 7) / 8, 256, 0, stream>>>(side, ego,
                                                          W_gc + (size_t)k * D * D, b_gc + (size_t)k * D,
                                                          W_bi + (size_t)k * D * D, b_bi + (size_t)k * D);
        k_accum<<<(2 * BATCH * D + 255) / 256, 256, 0, stream>>>(ego, users, items, uacc, iacc);
    }

    k_dot<<<(BATCH * 32 + 255) / 256, 256, 0, stream>>>(uacc, iacc, out);
}